// NodeGraph_18640158064651
// MI455X (gfx1250) — compile-verified
//
#include <hip/hip_runtime.h>
#include <math.h>

#define NODE  2048
#define DIM   64
#define SEQ   168
#define BATCH 128
#define HID   64
#define TOPK  20
#define KIN   (BATCH*SEQ)      /* 21504 */
#define ALPHA 3.0f
#define SLOPE 0.01f
#define BN_EPS 1e-5f

#define KSPLIT 16              /* split-K for layer0: 21504/16 = 1344 = 42*32 */
#define L0_KC  32
#define L0_NT  128
#define QB_JS  8               /* split over j for q@bank */
#define QB_KC  32
#define L2_NT  64
#define L2_JT  128
#define MX_IT  128
#define MX_JT  64

typedef __attribute__((ext_vector_type(2))) float v2f;
typedef __attribute__((ext_vector_type(8))) float v8f;

// Tensor Data Mover (gfx1250): this toolchain exposes the 6-arg builtin
// (uint32x4 g0, int32x8 g1, int32x4 g2, int32x4 g3, int32x8, i32 cpol).
#if defined(__HIP_DEVICE_COMPILE__) && \
    __has_builtin(__builtin_amdgcn_tensor_load_to_lds) && \
    __has_builtin(__builtin_amdgcn_s_wait_tensorcnt)
#define USE_TDM 1
#else
#define USE_TDM 0
#endif

typedef unsigned int u32x4 __attribute__((ext_vector_type(4)));
typedef int          i32x8 __attribute__((ext_vector_type(8)));
typedef int          i32x4 __attribute__((ext_vector_type(4)));

#if USE_TDM
// 2-D tiled TDM load: tile (tile_d1 rows x tile_d0 elems) of a (tensor_d1 x
// tensor_d0) fp32 tensor, row stride stride0 (elems), packed into LDS rows.
// D# bitfields per CDNA5 ISA 8.3/8.4 (group0: count/lds/global/type,
// group1: data_size, dims, tile dims, strides; groups 2/3/extra zero for 2-D).
__device__ __forceinline__ void tdm_load_2d(
    unsigned lds_off, const void* gaddr,
    unsigned tensor_d0, unsigned tensor_d1,
    unsigned tile_d0, unsigned tile_d1, unsigned stride0)
{
  unsigned long long ga = (unsigned long long)(size_t)gaddr;
  u32x4 g0;
  g0[0] = 1u;                                           // count=1, user desc
  g0[1] = lds_off;                                      // lds_addr (bytes)
  g0[2] = (unsigned)(ga & 0xFFFFFFFFu);                 // global_addr[31:0]
  g0[3] = (unsigned)((ga >> 32) & 0x01FFFFFFu)          // global_addr[56:32]
        | (2u << 30);                                   // type = 2 ("image")
  i32x8 g1;
  g1[0] = (int)(2u << 16);                              // data_size=4B, mask=0
  g1[1] = (int)((tensor_d0 & 0xFFFFu) << 16);           // tensor_dim0[15:0]
  g1[2] = (int)((tensor_d0 >> 16) |
                ((tensor_d1 & 0xFFFFu) << 16));         // dim0 hi | dim1 lo
  g1[3] = (int)((tensor_d1 >> 16) | (tile_d0 << 16));   // dim1 hi | tile_dim0
  g1[4] = (int)(tile_d1 & 0xFFFFu);                     // tile_dim1, tile_dim2=0
  g1[5] = (int)stride0;                                 // dim0_stride[31:0]
  g1[6] = 0;                                            // stride hi, dim1_stride
  g1[7] = 0;
  i32x4 gz4 = {0, 0, 0, 0};
  i32x8 gz8 = {0, 0, 0, 0, 0, 0, 0, 0};
  __builtin_amdgcn_tensor_load_to_lds(g0, g1, gz4, gz4, gz8, 0);
}
#endif

__device__ __forceinline__ v8f wmma_f32(v2f a, v2f b, v8f c) {
  // D = A(16x4, f32) x B(4x16, f32) + C(16x16, f32) -- exact fp32 matrix op
  return __builtin_amdgcn_wmma_f32_16x16x4_f32(false, a, false, b, (short)0, c,
                                               false, false);
}
__device__ __forceinline__ float lrelu(float x) { return x > 0.f ? x : SLOPE * x; }

// ---------------------------------------------------------------------------
// Layer 0: Z[m][n][h] partials.  A[k][n] = input_flat (k-major, coalesced in n).
// Fuses both weight sets (e1/e2) per input so lf/hf stream from HBM once.
// gfx1250 path: TDM double-buffered HBM->LDS DMA (TENSORcnt) overlapped with
// WMMA compute.  grid (NODE/128, KSPLIT, 2-inputs), block 256 (8 waves).
// ---------------------------------------------------------------------------
__global__ __launch_bounds__(256) void l0_gemm(
    const float* __restrict__ lowf, const float* __restrict__ highf,
    const float* __restrict__ w0_m0, const float* __restrict__ w0_m1,
    const float* __restrict__ w0_m2, const float* __restrict__ w0_m3,
    float* __restrict__ part)
{
  __shared__ float sA[2][L0_KC][L0_NT];      // 2 x 16 KB
  __shared__ float sW[2][2][HID][L0_KC];     // 2 x 16 KB
  const int tid = threadIdx.x;
  const int wave = tid >> 5, lane = tid & 31;
  const int half = lane >> 4, lm = lane & 15;
  const int n0 = blockIdx.x * L0_NT;
  const int ks = blockIdx.y;
  const int z  = blockIdx.z;
  const float* A = z ? highf : lowf;
  const float* W0 = z ? w0_m2 : w0_m0;
  const float* W1 = z ? w0_m3 : w0_m1;
  const int kbeg = ks * (KIN / KSPLIT);
  const int nch = (KIN / KSPLIT) / L0_KC;    // 42 chunks
  const int wm = wave * 16;

  v8f zero = {};
  v8f acc[2][4];
  for (int s = 0; s < 2; s++)
    for (int t = 0; t < 4; t++) acc[s][t] = zero;

#if USE_TDM
  // wave 0 -> A tile (32k x 128n), wave 1 -> W0 tile, wave 2 -> W1 tile
  auto issue = [&](int b, int k0) {
    if (wave == 0)
      tdm_load_2d((unsigned)(size_t)&sA[b][0][0],
                  A + (size_t)k0 * NODE + n0,
                  NODE, KIN, L0_NT, L0_KC, NODE);
    else if (wave == 1)
      tdm_load_2d((unsigned)(size_t)&sW[b][0][0][0],
                  W0 + (size_t)k0, KIN, HID, L0_KC, HID, KIN);
    else if (wave == 2)
      tdm_load_2d((unsigned)(size_t)&sW[b][1][0][0],
                  W1 + (size_t)k0, KIN, HID, L0_KC, HID, KIN);
  };
  issue(0, kbeg);
#endif

  for (int c = 0; c < nch; c++) {
    const int b = c & 1;
    const int k0 = kbeg + c * L0_KC;
#if USE_TDM
    if (wave < 3) __builtin_amdgcn_s_wait_tensorcnt(0);  // chunk c landed
    __syncthreads();                                     // visible to all waves
    if (c + 1 < nch) issue(b ^ 1, k0 + L0_KC);           // prefetch next chunk
#else
    __syncthreads();
    for (int i = tid; i < L0_KC * L0_NT; i += 256) {
      int kk = i >> 7, nn = i & (L0_NT - 1);
      sA[b][kk][nn] = A[(size_t)(k0 + kk) * NODE + (n0 + nn)];
    }
    for (int i = tid; i < 2 * HID * L0_KC; i += 256) {
      int s = i >> 11, r = i & 2047;
      int h = r >> 5, kk = r & 31;
      const float* W = s ? W1 : W0;
      sW[b][s][h][kk] = W[(size_t)h * KIN + (k0 + kk)];
    }
    __syncthreads();
#endif
#pragma unroll
    for (int kk = 0; kk < L0_KC; kk += 4) {
      v2f a;
      a.x = sA[b][kk + 2 * half + 0][wm + lm];
      a.y = sA[b][kk + 2 * half + 1][wm + lm];
#pragma unroll
      for (int s = 0; s < 2; s++) {
#pragma unroll
        for (int ht = 0; ht < 4; ht++) {
          v2f bb;
          bb.x = sW[b][s][ht * 16 + lm][kk + 2 * half + 0];
          bb.y = sW[b][s][ht * 16 + lm][kk + 2 * half + 1];
          acc[s][ht] = wmma_f32(a, bb, acc[s][ht]);
        }
      }
    }
#if USE_TDM
    __syncthreads();   // all waves done with buffer b before chunk c+2 lands
#endif
  }
  for (int s = 0; s < 2; s++) {
    float* P = part + (size_t)((z * KSPLIT + ks) * 2 + s) * NODE * HID;
#pragma unroll
    for (int ht = 0; ht < 4; ht++)
#pragma unroll
      for (int v = 0; v < 8; v++) {
        int node = n0 + wm + v + 8 * half;
        int h = ht * 16 + lm;
        P[(size_t)node * HID + h] = acc[s][ht][v];
      }
  }
}

// Deterministic split-K reduction (+bias).  m: 0=e1_qlow 1=e2_qlow 2=e1_qhigh 3=e2_qhigh
__global__ __launch_bounds__(256) void l0_reduce(
    const float* __restrict__ part, float* __restrict__ Z,
    const float* __restrict__ b_m0, const float* __restrict__ b_m1,
    const float* __restrict__ b_m2, const float* __restrict__ b_m3)
{
  int idx = blockIdx.x * 256 + threadIdx.x;
  if (idx >= 4 * NODE * HID) return;
  int m = idx >> 17;
  int inner = idx & (NODE * HID - 1);
  int h = idx & (HID - 1);
  int z = m >> 1, s = m & 1;
  float sum = 0.f;
  for (int ks = 0; ks < KSPLIT; ks++)
    sum += part[(size_t)((z * KSPLIT + ks) * 2 + s) * NODE * HID + inner];
  const float* b = (m == 0) ? b_m0 : (m == 1) ? b_m1 : (m == 2) ? b_m2 : b_m3;
  Z[idx] = sum + b[h];
}

// Training-mode BN column stats (biased var over 2048 rows) -> scale/shift
__global__ __launch_bounds__(256) void bn_stats(
    const float* __restrict__ Z, float* __restrict__ stats,
    const float* g0, const float* g1, const float* g2, const float* g3,
    const float* e0, const float* e1, const float* e2, const float* e3)
{
  __shared__ float ssum[256], ssq[256];
  int m = blockIdx.x;
  const float* X = Z + (size_t)m * NODE * HID;
  int t = threadIdx.x;
  int h = t & (HID - 1);
  int p = t >> 6;
  float s = 0.f, q = 0.f;
  for (int r = p * (NODE / 4); r < (p + 1) * (NODE / 4); r++) {
    float x = X[(size_t)r * HID + h];
    s += x; q += x * x;
  }
  ssum[t] = s; ssq[t] = q;
  __syncthreads();
  if (t < HID) {
    float ts = 0.f, tq = 0.f;
    for (int pp = 0; pp < 4; pp++) { ts += ssum[pp * 64 + h]; tq += ssq[pp * 64 + h]; }
    float mean = ts / (float)NODE;
    float var  = tq / (float)NODE - mean * mean;
    const float* g  = (m == 0) ? g0 : (m == 1) ? g1 : (m == 2) ? g2 : g3;
    const float* be = (m == 0) ? e0 : (m == 1) ? e1 : (m == 2) ? e2 : e3;
    float scale = g[h] * rsqrtf(var + BN_EPS);
    stats[m * 2 * HID + h]       = scale;
    stats[m * 2 * HID + HID + h] = be[h] - mean * scale;
  }
}

__global__ __launch_bounds__(256) void bn_apply(float* __restrict__ Z,
                                                const float* __restrict__ stats)
{
  int idx = blockIdx.x * 256 + threadIdx.x;
  if (idx >= 4 * NODE * HID) return;
  int m = idx >> 17;
  int h = idx & (HID - 1);
  float x = Z[idx] * stats[m * 2 * HID + h] + stats[m * 2 * HID + HID + h];
  Z[idx] = lrelu(x);
}

// Layer 1: [2048,64] @ w1^T + b1, WMMA.  grid (NODE/128, 4)
__global__ __launch_bounds__(256) void l1_gemm(
    const float* __restrict__ Zin, float* __restrict__ Zout,
    const float* w_m0, const float* w_m1, const float* w_m2, const float* w_m3,
    const float* b_m0, const float* b_m1, const float* b_m2, const float* b_m3)
{
  __shared__ float sX[L0_NT][HID];   // 32 KB
  __shared__ float sW[HID][HID];     // 16 KB
  int tid = threadIdx.x, wave = tid >> 5, lane = tid & 31;
  int half = lane >> 4, lm = lane & 15;
  int n0 = blockIdx.x * L0_NT;
  int m  = blockIdx.y;
  const float* X = Zin + (size_t)m * NODE * HID;
  const float* W = (m == 0) ? w_m0 : (m == 1) ? w_m1 : (m == 2) ? w_m2 : w_m3;
  const float* B = (m == 0) ? b_m0 : (m == 1) ? b_m1 : (m == 2) ? b_m2 : b_m3;
  for (int i = tid; i < L0_NT * HID; i += 256) { int n = i >> 6, h = i & 63; sX[n][h] = X[(size_t)(n0 + n) * HID + h]; }
  for (int i = tid; i < HID * HID; i += 256)   { int hp = i >> 6, h = i & 63; sW[hp][h] = W[(size_t)hp * HID + h]; }
  __syncthreads();
  int wm = wave * 16;
  v8f zero = {}; v8f acc[4]; for (int t = 0; t < 4; t++) acc[t] = zero;
#pragma unroll
  for (int k = 0; k < HID; k += 4) {
    v2f a; a.x = sX[wm + lm][k + 2 * half]; a.y = sX[wm + lm][k + 2 * half + 1];
#pragma unroll
    for (int ht = 0; ht < 4; ht++) {
      v2f b; b.x = sW[ht * 16 + lm][k + 2 * half]; b.y = sW[ht * 16 + lm][k + 2 * half + 1];
      acc[ht] = wmma_f32(a, b, acc[ht]);
    }
  }
  float* O = Zout + (size_t)m * NODE * HID;
  for (int ht = 0; ht < 4; ht++)
    for (int v = 0; v < 8; v++) {
      int node = n0 + wm + v + 8 * half, h = ht * 16 + lm;
      O[(size_t)node * HID + h] = acc[ht][v] + B[h];
    }
}

// Layer 2: logits[n][j] = X[n,:].w2[j,:] + b2[j].  grid (NODE/64, NODE/128)
__global__ __launch_bounds__(256) void l2_gemm(
    const float* __restrict__ X, const float* __restrict__ W2,
    const float* __restrict__ B2, float* __restrict__ logits)
{
  __shared__ float sX[L2_NT][HID];   // 16 KB
  __shared__ float sW[L2_JT][HID];   // 32 KB
  int tid = threadIdx.x, wave = tid >> 5, lane = tid & 31;
  int half = lane >> 4, lm = lane & 15;
  int n0 = blockIdx.x * L2_NT;
  int j0 = blockIdx.y * L2_JT;
  for (int i = tid; i < L2_NT * HID; i += 256) { int n = i >> 6, h = i & 63; sX[n][h] = X[(size_t)(n0 + n) * HID + h]; }
  for (int i = tid; i < L2_JT * HID; i += 256) { int j = i >> 6, h = i & 63; sW[j][h] = W2[(size_t)(j0 + j) * HID + h]; }
  __syncthreads();
  int iw = (wave & 3) * 16;     // node strip within 64
  int jh = (wave >> 2) * 64;    // j half within 128
  v8f zero = {}; v8f acc[4]; for (int t = 0; t < 4; t++) acc[t] = zero;
#pragma unroll
  for (int k = 0; k < HID; k += 4) {
    v2f a; a.x = sX[iw + lm][k + 2 * half]; a.y = sX[iw + lm][k + 2 * half + 1];
#pragma unroll
    for (int jt = 0; jt < 4; jt++) {
      v2f b; b.x = sW[jh + jt * 16 + lm][k + 2 * half]; b.y = sW[jh + jt * 16 + lm][k + 2 * half + 1];
      acc[jt] = wmma_f32(a, b, acc[jt]);
    }
  }
  for (int jt = 0; jt < 4; jt++)
    for (int v = 0; v < 8; v++) {
      int n = n0 + iw + v + 8 * half;
      int j = j0 + jh + jt * 16 + lm;
      logits[(size_t)n * NODE + j] = acc[jt][v] + B2[j];
    }
}

// Row softmax in-place, one block per row
__global__ __launch_bounds__(256) void softmax_rows(float* __restrict__ logits)
{
  __shared__ float row[NODE];
  __shared__ float red[256];
  int r = blockIdx.x, t = threadIdx.x;
  float* L = logits + (size_t)r * NODE;
  float mx = -__builtin_inff();
  for (int c = t; c < NODE; c += 256) { float x = L[c]; row[c] = x; mx = fmaxf(mx, x); }
  red[t] = mx;
  __syncthreads();
  for (int s = 128; s > 0; s >>= 1) { if (t < s) red[t] = fmaxf(red[t], red[t + s]); __syncthreads(); }
  float M = red[0];
  __syncthreads();
  float sum = 0.f;
  for (int c = t; c < NODE; c += 256) { float e = expf(row[c] - M); row[c] = e; sum += e; }
  red[t] = sum;
  __syncthreads();
  for (int s = 128; s > 0; s >>= 1) { if (t < s) red[t] += red[t + s]; __syncthreads(); }
  float inv = 1.f / red[0];
  for (int c = t; c < NODE; c += 256) L[c] = row[c] * inv;
}

// q @ bank, split over j into QB_JS partials.  grid (NODE/128, QB_JS)
__global__ __launch_bounds__(256) void qbank_gemm(
    const float* __restrict__ q, const float* __restrict__ bank,
    float* __restrict__ partout)
{
  __shared__ float sQ[L0_NT][QB_KC];   // 16 KB
  __shared__ float sB[QB_KC][DIM];     // 8 KB
  int tid = threadIdx.x, wave = tid >> 5, lane = tid & 31;
  int half = lane >> 4, lm = lane & 15;
  int n0 = blockIdx.x * L0_NT;
  int js = blockIdx.y;
  int jbeg = js * (NODE / QB_JS);
  int wm = wave * 16;
  v8f zero = {}; v8f acc[4]; for (int t = 0; t < 4; t++) acc[t] = zero;
  for (int jb = jbeg; jb < jbeg + NODE / QB_JS; jb += QB_KC) {
    __syncthreads();
    for (int i = tid; i < L0_NT * QB_KC; i += 256) { int n = i >> 5, c = i & 31; sQ[n][c] = q[(size_t)(n0 + n) * NODE + jb + c]; }
    for (int i = tid; i < QB_KC * DIM; i += 256)  { int rr = i >> 6, d = i & 63; sB[rr][d] = bank[(size_t)(jb + rr) * DIM + d]; }
    __syncthreads();
#pragma unroll
    for (int kk = 0; kk < QB_KC; kk += 4) {
      v2f a; a.x = sQ[wm + lm][kk + 2 * half]; a.y = sQ[wm + lm][kk + 2 * half + 1];
#pragma unroll
      for (int dt = 0; dt < 4; dt++) {
        v2f b; b.x = sB[kk + 2 * half][dt * 16 + lm]; b.y = sB[kk + 2 * half + 1][dt * 16 + lm];
        acc[dt] = wmma_f32(a, b, acc[dt]);
      }
    }
  }
  float* P = partout + (size_t)js * NODE * DIM;
  for (int dt = 0; dt < 4; dt++)
    for (int v = 0; v < 8; v++) {
      int node = n0 + wm + v + 8 * half, d = dt * 16 + lm;
      P[(size_t)node * DIM + d] = acc[dt][v];
    }
}

// vec[e] (+)= 3*lrelu(sum_js partials)
__global__ __launch_bounds__(256) void qbank_combine(
    const float* __restrict__ partin, float* __restrict__ vec, int beta)
{
  int idx = blockIdx.x * 256 + threadIdx.x;
  if (idx >= NODE * DIM) return;
  float s = 0.f;
  for (int js = 0; js < QB_JS; js++) s += partin[(size_t)js * NODE * DIM + idx];
  float nl = 3.0f * lrelu(s);
  vec[idx] = beta ? (vec[idx] + nl) : nl;
}

// adj = relu(tanh(3*(v1 v2^T - v2 v1^T))), both terms fused.  grid (16, 32)
__global__ __launch_bounds__(256) void adj_kernel(
    const float* __restrict__ vec1, const float* __restrict__ vec2,
    float* __restrict__ out)
{
  int tid = threadIdx.x, wave = tid >> 5, lane = tid & 31;
  int half = lane >> 4, lm = lane & 15;
  int i0 = blockIdx.x * MX_IT + wave * 16;
  int j0 = blockIdx.y * MX_JT;
  v8f zero = {}; v8f accP[4], accN[4];
  for (int t = 0; t < 4; t++) { accP[t] = zero; accN[t] = zero; }
#pragma unroll
  for (int k = 0; k < DIM; k += 4) {
    int kr = k + 2 * half;
    v2f a1, a2;
    a1.x = vec1[(size_t)(i0 + lm) * DIM + kr]; a1.y = vec1[(size_t)(i0 + lm) * DIM + kr + 1];
    a2.x = vec2[(size_t)(i0 + lm) * DIM + kr]; a2.y = vec2[(size_t)(i0 + lm) * DIM + kr + 1];
#pragma unroll
    for (int jt = 0; jt < 4; jt++) {
      int jr = j0 + jt * 16 + lm;
      v2f b1, b2;
      b2.x = vec2[(size_t)jr * DIM + kr]; b2.y = vec2[(size_t)jr * DIM + kr + 1];
      b1.x = vec1[(size_t)jr * DIM + kr]; b1.y = vec1[(size_t)jr * DIM + kr + 1];
      accP[jt] = wmma_f32(a1, b2, accP[jt]);
      accN[jt] = wmma_f32(a2, b1, accN[jt]);
    }
  }
  for (int jt = 0; jt < 4; jt++)
    for (int v = 0; v < 8; v++) {
      int i = i0 + v + 8 * half;
      int j = j0 + jt * 16 + lm;
      float m = accP[jt][v] - accN[jt][v];
      float a = tanhf(ALPHA * m);
      out[(size_t)i * NODE + j] = a > 0.f ? a : 0.f;
    }
}

// Per-row top-20 of (adj+noise), deterministic lowest-index tie-break, in-place mask
__global__ __launch_bounds__(256) void topk_mask(float* __restrict__ out,
                                                 const float* __restrict__ noise)
{
  __shared__ float aval[NODE];   // 8 KB
  __shared__ float key[NODE];    // 8 KB
  __shared__ float rv[256];
  __shared__ int   ri[256];
  int r = blockIdx.x, t = threadIdx.x;
  float* O = out + (size_t)r * NODE;
  const float* Nz = noise + (size_t)r * NODE;
  for (int c = t; c < NODE; c += 256) { float a = O[c]; aval[c] = a; key[c] = a + Nz[c]; }
  __syncthreads();
  const float NEG = -__builtin_inff();
  for (int it = 0; it < TOPK; it++) {
    float bv = NEG; int bi = -1;
    for (int c = t; c < NODE; c += 256) {
      float v = key[c];
      if (v > bv) { bv = v; bi = c; }     // ascending scan keeps lowest index on ties
    }
    rv[t] = bv; ri[t] = bi;
    __syncthreads();
    for (int s = 128; s > 0; s >>= 1) {
      if (t < s) {
        bool take = (rv[t + s] > rv[t]) ||
                    (rv[t + s] == rv[t] && ri[t + s] >= 0 && (ri[t] < 0 || ri[t + s] < ri[t]));
        if (take) { rv[t] = rv[t + s]; ri[t] = ri[t + s]; }
      }
      __syncthreads();
    }
    if (t == 0 && ri[0] >= 0) key[ri[0]] = NEG;
    __syncthreads();
  }
  for (int c = t; c < NODE; c += 256) O[c] = (key[c] == NEG) ? aval[c] : 0.f;
}

// ---------------------------------------------------------------------------
extern "C" void kernel_launch(void* const* d_in, const int* in_sizes, int n_in,
                              void* d_out, int out_size, void* d_ws, size_t ws_size,
                              hipStream_t stream)
{
  (void)n_in; (void)out_size; (void)ws_size;
  struct QP { const float *w0,*b0,*g0,*be0,*w1,*b1,*g1,*be1,*w2,*b2; };
  auto F = [&](int i) { return (const float*)d_in[i]; };
  // Distinguish insertion-order vs jax-sorted leaf flattening via in_sizes[2]:
  // insertion -> e1_low_bank (131072); sorted -> noise (4194304)
  bool sorted_layout = (in_sizes[2] == NODE * NODE);
  auto loadQ = [&](int base) -> QP {
    QP p;
    if (!sorted_layout) {  // w0,b0,g0,be0,w1,b1,g1,be1,w2,b2
      p.w0 = F(base+0); p.b0 = F(base+1); p.g0 = F(base+2); p.be0 = F(base+3);
      p.w1 = F(base+4); p.b1 = F(base+5); p.g1 = F(base+6); p.be1 = F(base+7);
      p.w2 = F(base+8); p.b2 = F(base+9);
    } else {               // sorted: b0,b1,b2,be0,be1,g0,g1,w0,w1,w2
      p.b0 = F(base+0); p.b1 = F(base+1); p.b2 = F(base+2);
      p.be0 = F(base+3); p.be1 = F(base+4); p.g0 = F(base+5); p.g1 = F(base+6);
      p.w0 = F(base+7); p.w1 = F(base+8); p.w2 = F(base+9);
    }
    return p;
  };
  const float *lowf, *highf, *noise;
  const float *bank_e1_low, *bank_e1_high, *bank_e2_low, *bank_e2_high;
  QP q[4];  // m: 0=e1_qlow 1=e2_qlow 2=e1_qhigh 3=e2_qhigh
  if (!sorted_layout) {
    lowf = F(0); highf = F(1);
    bank_e1_low = F(2); bank_e1_high = F(3);
    q[0] = loadQ(4);  q[2] = loadQ(14);
    bank_e2_low = F(24); bank_e2_high = F(25);
    q[1] = loadQ(26); q[3] = loadQ(36);
    noise = F(46);
  } else {
    highf = F(0); lowf = F(1); noise = F(2);
    bank_e1_high = F(3); bank_e1_low = F(4);
    q[2] = loadQ(5);  q[0] = loadQ(15);
    bank_e2_high = F(25); bank_e2_low = F(26);
    q[3] = loadQ(27); q[1] = loadQ(37);
  }

  // Workspace layout (floats). part (33.5MB) is dead before logits (16MB) is
  // live, so they share the same region.  Total ~41 MB.
  float* ws     = (float*)d_ws;
  float* part   = ws;                       // 2*16*2*NODE*HID = 8,388,608
  float* logits = ws;                       // NODE*NODE       = 4,194,304
  float* Z      = ws + 8388608;             // 4*NODE*HID      =   524,288
  float* Z2     = ws + 8912896;             // 4*NODE*HID      =   524,288
  float* stats  = ws + 9437184;             // 4*2*HID         =       512
  float* vec    = ws + 9437696;             // 2*NODE*DIM      =   262,144
  float* qpart  = ws + 9699840;             // QB_JS*NODE*DIM  = 1,048,576
  float* adj    = (float*)d_out;

  dim3 b256(256);
  // MLP layer 0 (reads lf/hf from HBM exactly once each; split-K for occupancy)
  l0_gemm<<<dim3(NODE / L0_NT, KSPLIT, 2), b256, 0, stream>>>(
      lowf, highf, q[0].w0, q[1].w0, q[2].w0, q[3].w0, part);
  l0_reduce<<<dim3((4 * NODE * HID) / 256), b256, 0, stream>>>(
      part, Z, q[0].b0, q[1].b0, q[2].b0, q[3].b0);
  bn_stats<<<dim3(4), b256, 0, stream>>>(Z, stats,
      q[0].g0, q[1].g0, q[2].g0, q[3].g0, q[0].be0, q[1].be0, q[2].be0, q[3].be0);
  bn_apply<<<dim3((4 * NODE * HID) / 256), b256, 0, stream>>>(Z, stats);
  // MLP layer 1
  l1_gemm<<<dim3(NODE / L0_NT, 4), b256, 0, stream>>>(Z, Z2,
      q[0].w1, q[1].w1, q[2].w1, q[3].w1, q[0].b1, q[1].b1, q[2].b1, q[3].b1);
  bn_stats<<<dim3(4), b256, 0, stream>>>(Z2, stats,
      q[0].g1, q[1].g1, q[2].g1, q[3].g1, q[0].be1, q[1].be1, q[2].be1, q[3].be1);
  bn_apply<<<dim3((4 * NODE * HID) / 256), b256, 0, stream>>>(Z2, stats);
  // Layer 2 + softmax + q@bank, one q-matrix at a time (logits buffer reused)
  struct Stage { int m; const float* bank; int e; int beta; };
  Stage st[4] = { {0, bank_e1_low, 0, 0}, {2, bank_e1_high, 0, 1},
                  {1, bank_e2_low, 1, 0}, {3, bank_e2_high, 1, 1} };
  for (int si = 0; si < 4; si++) {
    int m = st[si].m;
    l2_gemm<<<dim3(NODE / L2_NT, NODE / L2_JT), b256, 0, stream>>>(
        Z2 + (size_t)m * NODE * HID, q[m].w2, q[m].b2, logits);
    softmax_rows<<<dim3(NODE), b256, 0, stream>>>(logits);
    qbank_gemm<<<dim3(NODE / L0_NT, QB_JS), b256, 0, stream>>>(
        logits, st[si].bank, qpart);
    qbank_combine<<<dim3((NODE * DIM) / 256), b256, 0, stream>>>(
        qpart, vec + (size_t)st[si].e * NODE * DIM, st[si].beta);
  }
  // Antisymmetric outer product + relu(tanh(3x)) -> d_out
  adj_kernel<<<dim3(NODE / MX_IT, NODE / MX_JT), b256, 0, stream>>>(
      vec, vec + (size_t)NODE * DIM, adj);
  // Per-row top-20 mask (in-place)
  topk_mask<<<dim3(NODE), b256, 0, stream>>>(adj, noise);
}